// TPS_1279900254572
// MI455X (gfx1250) — compile-verified
//
#include <hip/hip_runtime.h>
#include <hip/hip_bf16.h>

// TPS forward for gfx1250 (MI455X).
//   out[b,p,c] = sum_{k=0..52} phi_k(p) * param[b][k][c]
//   phi_k(p) = r2*log(r2+eps) for k<50 (RBF), x for k=50, y for k=51, 1 for k=52
//   param = solve(L, Y) with Y == 0 -> param == 0 -> out == 0 (still done generally).
//
// Kernel 1: per-batch 53x53 Gauss-Jordan solve in LDS -> param in d_ws.
// Kernel 2: grid eval; (16px x 53basis) x (53 x 2) contraction via
//           V_WMMA_F32_16X16X4_F32 (K padded to 56, N padded to 16).
//           kp/param staged to LDS via global_load_async_to_lds when available.

#define TPS_EPS 1e-9f
#define TPS_REG 0.01f
#define TPS_BS  8
#define TPS_H   256
#define TPS_W   256
#define TPS_NKP 50
#define TPS_NB  53        // 50 RBF + x + y + 1
#define TPS_KSTEPS 14     // ceil(53/4) K-steps of 4 (K padded to 56)
#define TPS_RBFSTEPS 12   // steps whose basis indices are all < 50 for every lane
#define TPS_PWSTRIDE 112  // padded param stride per batch in workspace (floats)

typedef float v2f __attribute__((ext_vector_type(2)));
typedef float v8f __attribute__((ext_vector_type(8)));

// r2 * ln(r2 + eps). Argument of the log is always normal/finite/positive
// (>= 1e-9), so the raw v_log_f32 (log2) plus an ln2 fold is exact enough:
// r2*ln(x) = (r2*ln2) * log2(x).  Avoids the ~14-op precise-log expansion.
__device__ __forceinline__ float tps_u(float r2) {
#if __has_builtin(__builtin_amdgcn_logf)
    return (r2 * 0.69314718055994530942f) * __builtin_amdgcn_logf(r2 + TPS_EPS);
#else
    return r2 * __logf(r2 + TPS_EPS);
#endif
}

// ---------------------------------------------------------------------------
// Kernel 1: build L (53x53) per batch and Gauss-Jordan solve L * param = 0.
// One block (64 threads = 2 waves) per batch; everything lives in LDS.
// ---------------------------------------------------------------------------
__global__ __launch_bounds__(64) void tps_solve_kernel(const float* __restrict__ kp,
                                                       float* __restrict__ param_ws) {
    const int b = blockIdx.x;
    const int t = threadIdx.x;

    __shared__ float L[TPS_NB][56];   // padded row stride
    __shared__ float R[TPS_NB][2];
    __shared__ float kx[TPS_NKP], ky[TPS_NKP];

    if (t < TPS_NKP) {
        // jnp.where(kp == -1, -1.0, kp) is a numeric no-op for f32.
        kx[t] = kp[b * 2 * TPS_NKP + 2 * t + 0];
        ky[t] = kp[b * 2 * TPS_NKP + 2 * t + 1];
    }
    __syncthreads();

    for (int idx = t; idx < TPS_NB * TPS_NB; idx += 64) {
        const int i = idx / TPS_NB;
        const int j = idx % TPS_NB;
        float v;
        if (i < TPS_NKP && j < TPS_NKP) {
            const float dx = kx[i] - kx[j];
            const float dy = ky[i] - ky[j];
            v = tps_u(dx * dx + dy * dy);
        } else if (i < TPS_NKP) {           // columns 50..52: [x_i, y_i, 1]
            v = (j == 50) ? kx[i] : ((j == 51) ? ky[i] : 1.0f);
        } else if (j < TPS_NKP) {           // rows 50..52: [x_j, y_j, 1]
            v = (i == 50) ? kx[j] : ((i == 51) ? ky[j] : 1.0f);
        } else {
            v = 0.0f;
        }
        if (i == j) v += TPS_REG;
        L[i][j] = v;
    }
    if (t < TPS_NB * 2) R[t >> 1][t & 1] = 0.0f;   // Y == 0
    __syncthreads();

    // Gauss-Jordan (no pivoting; diagonal regularized). RHS==0 -> exact zeros.
    for (int p = 0; p < TPS_NB; ++p) {
        const float piv = L[p][p];
        __syncthreads();                   // everyone read the pivot before row scale
        const float inv = 1.0f / piv;
        for (int j = t; j < TPS_NB; j += 64) L[p][j] *= inv;
        if (t < 2) R[p][t] *= inv;
        __syncthreads();
        if (t < TPS_NB && t != p) {
            const float f = L[t][p];
            #pragma unroll 1
            for (int j = 0; j < TPS_NB; ++j) L[t][j] -= f * L[p][j];
            R[t][0] -= f * R[p][0];
            R[t][1] -= f * R[p][1];
        }
        __syncthreads();
    }

    float* pw = param_ws + (size_t)b * TPS_PWSTRIDE;
    if (t < TPS_NB * 2) pw[t] = R[t >> 1][t & 1];
    if (t >= TPS_NB * 2 && t < TPS_PWSTRIDE) pw[t] = 0.0f;   // K padding -> 0
}

// ---------------------------------------------------------------------------
// Kernel 2: grid evaluation with V_WMMA_F32_16X16X4_F32.
// grid = (32, 8); block = 256 (8 waves); each wave: 16 tiles x 16 pixels.
//   A (16x4 f32):  lane l: M = l&15, K = 4*step + 2*(l>>4) + {0,1}
//   B (4x16 f32):  lane l: N = l&15, K = 4*step + 2*(l>>4) + {0,1}  (hoisted)
//   C (16x16 f32): VGPR r, lane l: M = r + 8*(l>>4), N = l&15
// Steps 0..11 are all-RBF for every lane -> keypoints hoisted into VGPRs.
// ---------------------------------------------------------------------------
__global__ __launch_bounds__(256) void tps_eval_kernel(const float* __restrict__ kp,
                                                       const float* __restrict__ param_ws,
                                                       float* __restrict__ out) {
    const int b    = blockIdx.y;
    const int t    = threadIdx.x;
    const int lane = t & 31;
    const int wid  = t >> 5;

    __shared__ float s_kp[2 * TPS_NKP];   // interleaved x,y
    __shared__ float s_w[TPS_PWSTRIDE];
    __shared__ float s_stage[8 * 32];     // per-wave 16px x 2ch staging

#if __has_builtin(__builtin_amdgcn_global_load_async_to_lds_b32)
    {
        typedef __attribute__((address_space(1))) int glb_i;
        typedef __attribute__((address_space(3))) int lds_i;
        if (t < 2 * TPS_NKP)
            __builtin_amdgcn_global_load_async_to_lds_b32(
                (glb_i*)(unsigned long long)(kp + b * 2 * TPS_NKP + t),
                (lds_i*)(unsigned)(unsigned long long)&s_kp[t], 0, 0);
        if (t < TPS_PWSTRIDE)
            __builtin_amdgcn_global_load_async_to_lds_b32(
                (glb_i*)(unsigned long long)(param_ws + (size_t)b * TPS_PWSTRIDE + t),
                (lds_i*)(unsigned)(unsigned long long)&s_w[t], 0, 0);
#if __has_builtin(__builtin_amdgcn_s_wait_asynccnt)
        __builtin_amdgcn_s_wait_asynccnt(0);
#else
        asm volatile("s_wait_asynccnt 0x0" ::: "memory");
#endif
    }
#else
    if (t < 2 * TPS_NKP) s_kp[t] = kp[b * 2 * TPS_NKP + t];
    if (t < TPS_PWSTRIDE) s_w[t] = param_ws[(size_t)b * TPS_PWSTRIDE + t];
#endif
    __syncthreads();

    const int n  = lane & 15;          // N column for B/C fragments
    const int kk = (lane >> 4) << 1;   // K sub-offset: 0 or 2
    const int m  = lane & 15;          // pixel-in-tile (A-fragment M)

    // Hoisted B fragments (param matrix, zero-padded to 56 x 16).
    v2f Bf[TPS_KSTEPS];
    #pragma unroll
    for (int s = 0; s < TPS_KSTEPS; ++s) {
        const int k0 = 4 * s + kk;
        Bf[s].x = (n < 2 && k0     < TPS_NB) ? s_w[(k0    ) * 2 + n] : 0.0f;
        Bf[s].y = (n < 2 && k0 + 1 < TPS_NB) ? s_w[(k0 + 1) * 2 + n] : 0.0f;
    }

    // Hoisted keypoints for the pure-RBF K-steps (k = 4s+kk, 4s+kk+1 < 50).
    v2f akx[TPS_RBFSTEPS], aky[TPS_RBFSTEPS];
    #pragma unroll
    for (int s = 0; s < TPS_RBFSTEPS; ++s) {
        const int k0 = 4 * s + kk;
        akx[s].x = s_kp[2 * k0 + 0];  aky[s].x = s_kp[2 * k0 + 1];
        akx[s].y = s_kp[2 * k0 + 2];  aky[s].y = s_kp[2 * k0 + 3];
    }

    auto basis = [&](int k, float xg, float yg) -> float {
        if (k < TPS_NKP) {
            const float dx = xg - s_kp[2 * k];
            const float dy = yg - s_kp[2 * k + 1];
            return tps_u(dx * dx + dy * dy);
        }
        if (k == 50) return xg;
        if (k == 51) return yg;
        if (k == 52) return 1.0f;
        return 0.0f;
    };

    const int ITERS = 16;   // 32 blk * 8 waves * 16 it * 16 px = 65536
    float* st = &s_stage[wid * 32];

    for (int it = 0; it < ITERS; ++it) {
        const int tile = (blockIdx.x * 8 + wid) * ITERS + it;
        const int base = tile * 16;
        const int p    = base + m;
        const float xg = 2.0f * (float)(p & (TPS_W - 1)) * (1.0f / (TPS_W - 1)) - 1.0f;
        const float yg = 2.0f * (float)(p >> 8)          * (1.0f / (TPS_H - 1)) - 1.0f;

        v8f acc = {0.f, 0.f, 0.f, 0.f, 0.f, 0.f, 0.f, 0.f};

        #pragma unroll
        for (int s = 0; s < TPS_RBFSTEPS; ++s) {        // pure-RBF steps, regs only
            v2f A;
            const float dx0 = xg - akx[s].x, dy0 = yg - aky[s].x;
            A.x = tps_u(dx0 * dx0 + dy0 * dy0);
            const float dx1 = xg - akx[s].y, dy1 = yg - aky[s].y;
            A.y = tps_u(dx1 * dx1 + dy1 * dy1);
            acc = __builtin_amdgcn_wmma_f32_16x16x4_f32(
                false, A, false, Bf[s], (short)0, acc, false, false);
        }
        #pragma unroll
        for (int s = TPS_RBFSTEPS; s < TPS_KSTEPS; ++s) {  // mixed RBF/affine tail
            const int k0 = 4 * s + kk;
            v2f A;
            A.x = basis(k0,     xg, yg);
            A.y = basis(k0 + 1, xg, yg);
            acc = __builtin_amdgcn_wmma_f32_16x16x4_f32(
                false, A, false, Bf[s], (short)0, acc, false, false);
        }

        // Stage columns N=0,1 of the C tile into LDS, then coalesced store:
        // lane l stores out[b][base + l/2][l&1] -> one contiguous 128B line/wave.
        if (n < 2) {
            const int mb = (lane >> 4) * 8;
            #pragma unroll
            for (int r = 0; r < 8; ++r) st[(mb + r) * 2 + n] = acc[r];
        }
        asm volatile("s_wait_dscnt 0x0" ::: "memory");   // in-wave LDS RAW fence
        const float val = st[lane];
        out[(size_t)b * (TPS_H * TPS_W * 2) + (size_t)base * 2 + lane] = val;
    }
}

// ---------------------------------------------------------------------------
extern "C" void kernel_launch(void* const* d_in, const int* in_sizes, int n_in,
                              void* d_out, int out_size, void* d_ws, size_t ws_size,
                              hipStream_t stream) {
    // d_in[0] = source_image (8,3,256,256) f32 -- only its shape matters.
    // d_in[1] = kp_driving   (8,50,2)     f32
    const float* kp  = (const float*)d_in[1];
    float*       out = (float*)d_out;
    float*       pws = (float*)d_ws;     // 8 * 112 floats = 3584 B of scratch

    tps_solve_kernel<<<dim3(TPS_BS), dim3(64), 0, stream>>>(kp, pws);
    tps_eval_kernel<<<dim3(32, TPS_BS), dim3(256), 0, stream>>>(kp, pws, out);
}